// MLP_2D_InteriorMaterial_16939351015584
// MI455X (gfx1250) — compile-verified
//
#include <hip/hip_runtime.h>

typedef float v2f __attribute__((ext_vector_type(2)));
typedef float v8f __attribute__((ext_vector_type(8)));

// ---------------- LDS layout (in floats) ----------------
// Weights, zero-padded so that out-of-range rows/cols are exactly 0.
constexpr int LW1  = 0;              // 4  x 32  (2->30, K padded to 4)
constexpr int LB1  = LW1  + 4 * 32;  // 32
constexpr int LW2  = LB1  + 32;      // 32 x 32  (30->30)
constexpr int LB2  = LW2  + 32 * 32; // 32
constexpr int LW3  = LB2  + 32;      // 32 x 32
constexpr int LB3  = LW3  + 32 * 32; // 32
constexpr int LW4  = LB3  + 32;      // 32 x 16  (30->8)
constexpr int LB4  = LW4  + 32 * 16; // 16
constexpr int LP1  = LB4  + 16;      // 4  x 16  (2->15)
constexpr int LPB1 = LP1  + 4 * 16;  // 16
constexpr int LP2  = LPB1 + 16;      // 16 x 16  (15->15)
constexpr int LPB2 = LP2  + 16 * 16; // 16
constexpr int LP3  = LPB2 + 16;      // 16 x 16
constexpr int LPB3 = LP3  + 16 * 16; // 16
constexpr int LP4  = LPB3 + 16;      // 16 x 16  (15->4)
constexpr int LPB4 = LP4  + 16 * 16; // 16
constexpr int WTOT = LPB4 + 16;      // = 3696 floats

// Per-wave scratch region
constexpr int R_H    = 0;            // 16 x 32 activation tile
constexpr int R_XW   = R_H  + 512;   // 32 x 4  staged x   (cols 2,3 = 0)
constexpr int R_XB   = R_XW + 128;   // 32 x 4  staged x_ba
constexpr int R_WO   = R_XB + 128;   // 32 x 8  w-MLP outputs
constexpr int R_PO   = R_WO + 256;   // 32 x 4  phi-MLP outputs
constexpr int RSZ    = R_PO + 128;   // 1152 floats per wave
constexpr int NWAVES = 8;
constexpr int SMEM_F = WTOT + NWAVES * RSZ;   // 12912 floats = 51648 B

// ---- branchless transcendentals on the gfx1250 TRANS pipe ----
__device__ __forceinline__ float fast_tanh(float x) {
#if __has_builtin(__builtin_amdgcn_tanhf)
  return __builtin_amdgcn_tanhf(x);          // v_tanh_f32
#else
  const float ax = fabsf(x);
  // exp(-2|x|) = exp2(-2*log2(e)*|x|) via hardware v_exp_f32
  const float e = __builtin_amdgcn_exp2f(ax * -2.8853900817779268f);
  const float t = (1.0f - e) * __builtin_amdgcn_rcpf(1.0f + e);
  return copysignf(t, x);
#endif
}

__device__ __forceinline__ float fast_pow(float r, float p) {
  // r >= 0; r == 0 -> log2 = -inf -> exp2 = 0 (correct for p > 0)
  return __builtin_amdgcn_exp2f(p * __builtin_amdgcn_logf(r));
}

// One MLP layer on a 16-point tile via V_WMMA_F32_16X16X4_F32.
// A layout (16x4 fp32): lanes 0-15 hold K=4k,4k+1; lanes 16-31 hold K=4k+2,4k+3.
// B layout (4x16 fp32): VGPR0/1 hold rows kb,kb+1 (kb = 4k + 2*half), N = lane%16.
// D layout: element j -> row = j + 8*half, col = lane%16 (+16*nblock).
template <int KCH, int NBL, int LDIN, int LDW, int LDOUT, int SCOLS, bool ACT>
__device__ __forceinline__ void mlp_layer(const float* Ain, const float* W,
                                          const float* Bias, float* Aout,
                                          int l16, int half) {
  v8f acc[NBL];
#pragma unroll
  for (int nb = 0; nb < NBL; ++nb) {
    v8f c = {0.f, 0.f, 0.f, 0.f, 0.f, 0.f, 0.f, 0.f};
#pragma unroll
    for (int k = 0; k < KCH; ++k) {
      const int kb = 4 * k + 2 * half;
      v2f a;
      a.x = Ain[l16 * LDIN + kb];
      a.y = Ain[l16 * LDIN + kb + 1];
      const int n = l16 + 16 * nb;
      v2f b;
      b.x = W[kb * LDW + n];
      b.y = W[(kb + 1) * LDW + n];
      c = __builtin_amdgcn_wmma_f32_16x16x4_f32(false, a, false, b, (short)0, c,
                                                false, false);
    }
    acc[nb] = c;
  }
  __builtin_amdgcn_wave_barrier();   // keep all A reads before in-place stores
#pragma unroll
  for (int nb = 0; nb < NBL; ++nb) {
    const int col = l16 + 16 * nb;
    if (col < SCOLS) {
      const float bv = Bias[col];
#pragma unroll
      for (int j = 0; j < 8; ++j) {
        const int row = j + 8 * half;
        float v = acc[nb][j] + bv;
        Aout[row * LDOUT + col] = ACT ? fast_tanh(v) : v;
      }
    }
  }
  __builtin_amdgcn_wave_barrier();
}

__device__ __forceinline__ void copy_pad(const float* g, float* l, int rows,
                                         int cols, int ldd, int tid) {
  for (int i = tid; i < rows * cols; i += 256)
    l[(i / cols) * ldd + (i % cols)] = g[i];
}
__device__ __forceinline__ void copy_vec(const float* g, float* l, int n, int tid) {
  for (int i = tid; i < n; i += 256) l[i] = g[i];
}

__global__ __launch_bounds__(256) void zsf_mlp_kernel(
    const float* __restrict__ X, const float* __restrict__ IMV,
    const float* __restrict__ LMBD,
    const float* Ww1, const float* bw1, const float* Ww2, const float* bw2,
    const float* Ww3, const float* bw3, const float* Ww4, const float* bw4,
    const float* Wp1, const float* bp1, const float* Wp2, const float* bp2,
    const float* Wp3, const float* bp3, const float* Wp4, const float* bp4,
    float* __restrict__ out, int N) {
  __shared__ float smem[SMEM_F];
  const int tid  = threadIdx.x;
  const int wave = tid >> 5;
  const int lane = tid & 31;
  const int half = lane >> 4;
  const int l16  = lane & 15;

  // ---- stage weights into LDS (zero-padded) ----
  for (int i = tid; i < WTOT; i += 256) smem[i] = 0.f;
  __syncthreads();
  copy_pad(Ww1, smem + LW1, 2, 30, 32, tid);  copy_vec(bw1, smem + LB1, 30, tid);
  copy_pad(Ww2, smem + LW2, 30, 30, 32, tid); copy_vec(bw2, smem + LB2, 30, tid);
  copy_pad(Ww3, smem + LW3, 30, 30, 32, tid); copy_vec(bw3, smem + LB3, 30, tid);
  copy_pad(Ww4, smem + LW4, 30, 8, 16, tid);  copy_vec(bw4, smem + LB4, 8, tid);
  copy_pad(Wp1, smem + LP1, 2, 15, 16, tid);  copy_vec(bp1, smem + LPB1, 15, tid);
  copy_pad(Wp2, smem + LP2, 15, 15, 16, tid); copy_vec(bp2, smem + LPB2, 15, tid);
  copy_pad(Wp3, smem + LP3, 15, 15, 16, tid); copy_vec(bp3, smem + LPB3, 15, tid);
  copy_pad(Wp4, smem + LP4, 15, 4, 16, tid);  copy_vec(bp4, smem + LPB4, 4, tid);
  __syncthreads();

  float* ws = smem + WTOT + wave * RSZ;
  float* H  = ws + R_H;
  float* XW = ws + R_XW;
  float* XB = ws + R_XB;
  float* WO = ws + R_WO;
  float* PO = ws + R_PO;

  const float imv0 = IMV[0], imv1 = IMV[1];
  const float lm   = LMBD[0];

  const long long stride = (long long)gridDim.x * (NWAVES * 32);
  for (long long base = ((long long)blockIdx.x * NWAVES + wave) * 32; base < N;
       base += stride) {
    const long long p  = base + lane;
    const long long pc = (p < N) ? p : (long long)N - 1;
    __builtin_prefetch(X + 2 * (base + stride), 0, 0);

    const float x0 = X[2 * pc + 0];
    const float x1 = X[2 * pc + 1];
    const float d0 = x0 - imv0;
    const float d1 = x1 - imv1;
    const float r  = sqrtf(d0 * d0 + d1 * d1);
    const float ir = 1.0f / r;

    XW[lane * 4 + 0] = x0;      XW[lane * 4 + 1] = x1;
    XW[lane * 4 + 2] = 0.f;     XW[lane * 4 + 3] = 0.f;
    XB[lane * 4 + 0] = d0 * ir; XB[lane * 4 + 1] = d1 * ir;
    XB[lane * 4 + 2] = 0.f;     XB[lane * 4 + 3] = 0.f;
    __builtin_amdgcn_wave_barrier();

#pragma unroll
    for (int t = 0; t < 2; ++t) {
      // ---- w-MLP: 2 -> 30 -> 30 -> 30 -> 8 ----
      mlp_layer<1, 2, 4, 32, 32, 32, true >(XW + t * 64, smem + LW1, smem + LB1, H, l16, half);
      mlp_layer<8, 2, 32, 32, 32, 32, true >(H, smem + LW2, smem + LB2, H, l16, half);
      mlp_layer<8, 2, 32, 32, 32, 32, true >(H, smem + LW3, smem + LB3, H, l16, half);
      mlp_layer<8, 1, 32, 16, 8, 8, false>(H, smem + LW4, smem + LB4, WO + t * 128, l16, half);
      // ---- phi-MLP: 2 -> 15 -> 15 -> 15 -> 4 ----
      mlp_layer<1, 1, 4, 16, 32, 16, true >(XB + t * 64, smem + LP1, smem + LPB1, H, l16, half);
      mlp_layer<4, 1, 32, 16, 32, 16, true >(H, smem + LP2, smem + LPB2, H, l16, half);
      mlp_layer<4, 1, 32, 16, 32, 16, true >(H, smem + LP3, smem + LPB3, H, l16, half);
      mlp_layer<4, 1, 32, 16, 4, 4, false>(H, smem + LP4, smem + LPB4, PO + t * 64, l16, half);
    }

    // ---- scalar tail: one point per lane ----
    float w[8];
#pragma unroll
    for (int j = 0; j < 8; ++j) w[j] = WO[lane * 8 + j];
    const float ph0 = PO[lane * 4 + 0];
    const float ph1 = PO[lane * 4 + 1];
    const float ph2 = PO[lane * 4 + 2];
    const float ph3 = PO[lane * 4 + 3];
    __builtin_amdgcn_wave_barrier();

    // varphi(x): r0^k sin(k*theta), k = 0.5, 1, 1.5
    const float r0  = sqrtf(x0 * x0 + x1 * x1);
    const float th  = atan2f(x1, x0);
    const float sq  = sqrtf(r0);
    const float v0  = sq * sinf(0.5f * th);
    const float v1  = r0 * sinf(th);
    const float v2  = r0 * sq * sinf(1.5f * th);
    // varphi(x_ba): |x_ba| == 1 -> just sin(k * theta_d)
    const float thb = atan2f(d1, d0);
    const float s0  = sinf(0.5f * thb);
    const float s1  = sinf(thb);
    const float s2  = sinf(1.5f * thb);
    // yita(r): smoothstep-complement
    float tt = 2.5f * r - 1.25f;
    tt = fminf(fmaxf(tt, 0.f), 1.f);
    const float y = ((-6.f * tt + 15.f) * tt - 10.f) * (tt * tt * tt) + 1.f;

    const float rp = w[0] + y * w[4] + (w[1] + w[5] * y) * v0 +
                     (w[2] + w[6] * y) * v1 + (w[3] + w[7] * y) * v2;
    const float s  = ph0 + ph1 * s0 + ph2 * s1 + ph3 * s2;
    const float sp = s * y * fast_pow(r, lm);

    if (p < N) out[p] = rp + sp;
  }
}

extern "C" void kernel_launch(void* const* d_in, const int* in_sizes, int n_in,
                              void* d_out, int out_size, void* d_ws, size_t ws_size,
                              hipStream_t stream) {
  const int N = in_sizes[0] / 2;
  if (N <= 0) return;
  const float* X    = (const float*)d_in[0];
  const float* IMV  = (const float*)d_in[1];
  const float* LMBD = (const float*)d_in[2];
  const float* Ww1 = (const float*)d_in[3];  const float* bw1 = (const float*)d_in[4];
  const float* Ww2 = (const float*)d_in[5];  const float* bw2 = (const float*)d_in[6];
  const float* Ww3 = (const float*)d_in[7];  const float* bw3 = (const float*)d_in[8];
  const float* Ww4 = (const float*)d_in[9];  const float* bw4 = (const float*)d_in[10];
  const float* Wp1 = (const float*)d_in[11]; const float* bp1 = (const float*)d_in[12];
  const float* Wp2 = (const float*)d_in[13]; const float* bp2 = (const float*)d_in[14];
  const float* Wp3 = (const float*)d_in[15]; const float* bp3 = (const float*)d_in[16];
  const float* Wp4 = (const float*)d_in[17]; const float* bp4 = (const float*)d_in[18];

  int grid = (N + 255) / 256;
  if (grid > 4096) grid = 4096;
  zsf_mlp_kernel<<<grid, 256, 0, stream>>>(
      X, IMV, LMBD, Ww1, bw1, Ww2, bw2, Ww3, bw3, Ww4, bw4,
      Wp1, bp1, Wp2, bp2, Wp3, bp3, Wp4, bp4, (float*)d_out, N);
}